// GlobalAttentionPool_22849226015148
// MI455X (gfx1250) — compile-verified
//
#include <hip/hip_runtime.h>
#include <hip/hip_bf16.h>

#define NN  500000
#define BB  128
#define DD  256
#define HH  8
#define DHH 32
#define WIN 1024

typedef __attribute__((ext_vector_type(2))) float v2f;
typedef __attribute__((ext_vector_type(8))) float v8f;

// workspace layout (float offsets)
#define OFF_BPACK   0         // 4096   : wks packed in WMMA-B lane layout
#define OFF_BKS     4096      // 16     : folded bk (scaled), padded to 16
#define OFF_SEGMAX  4112      // 1024   : [B][8]
#define OFF_INV     5136      // 1024   : [B][8] 1/denom (or 1)
#define OFF_FLAG    6160      // 128    : [B] nonempty (1.0/0.0)
#define OFF_POOLEDX 8192      // 262144 : [B][8][256]
#define OFF_POOLED  270336    // 32768  : [B][256]
#define OFF_SCORES  303104    // 4000000: [N][8]

// ---------------- K0: fold q and Wk into a 256x16 score matrix ----------------
__global__ void k0_prep(const float* __restrict__ gt, const float* __restrict__ Wq,
                        const float* __restrict__ bq, const float* __restrict__ Wk,
                        const float* __restrict__ bk, float* __restrict__ ws) {
  __shared__ float s_gt[DD];
  __shared__ float s_q[DD];
  __shared__ float s_wks[DD * 16];
  int t = threadIdx.x;
  s_gt[t] = gt[t];
  __syncthreads();
  float acc = bq[t];
  for (int d = 0; d < DD; ++d) acc += s_gt[d] * Wq[d * DD + t];
  s_q[t] = acc;                                   // q[t]
  __syncthreads();
  const float scale = 0.1767766952966369f;        // 1/sqrt(32)
  // wks[k=t][c] = scale * sum_dh Wk[t][c*32+dh] * q[c*32+dh]; cols 8..15 zero pad
  for (int c = 0; c < 8; ++c) {
    float s = 0.f;
    for (int dh = 0; dh < DHH; ++dh)
      s += Wk[t * DD + c * DHH + dh] * s_q[c * DHH + dh];
    s_wks[t * 16 + c] = s * scale;
    s_wks[t * 16 + c + 8] = 0.f;
  }
  __syncthreads();
  // pack row k=t into WMMA B layout: element(k,c) -> flat ((k/4)*32 + ((k>>1)&1)*16 + c)*2 + (k&1)
  int k = t;
  for (int c = 0; c < 16; ++c) {
    int idx = (((k >> 2) * 32) + (((k >> 1) & 1) * 16) + c) * 2 + (k & 1);
    ws[OFF_BPACK + idx] = s_wks[k * 16 + c];
  }
  if (t < 8) {
    float s = 0.f;
    for (int dh = 0; dh < DHH; ++dh) s += bk[t * DHH + dh] * s_q[t * DHH + dh];
    ws[OFF_BKS + t] = s * scale;
  } else if (t < 16) {
    ws[OFF_BKS + t] = 0.f;
  }
}

// ---------------- K1: scores[N][8] = x @ wks + bks  (WMMA f32 16x16x4) --------
__global__ void k1_scores(const float* __restrict__ x, float* __restrict__ ws) {
  __shared__ v2f sB[2048];       // 64 k-steps x 32 lanes, pre-packed B operands
  __shared__ float s_bks[16];
  int t = threadIdx.x;
  const v2f* gB = (const v2f*)(ws + OFF_BPACK);
  for (int i = t; i < 2048; i += 256) sB[i] = gB[i];
  if (t < 16) s_bks[t] = ws[OFF_BKS + t];
  __syncthreads();
  int wave = t >> 5, lane = t & 31;
  int tile = blockIdx.x * 8 + wave;               // 16 nodes per wave
  if (tile * 16 >= NN) return;                    // wave-uniform guard (EXEC all 1s for WMMA)
  int m = lane & 15, lh = lane >> 4;
  const float* arow = x + (tile * 16 + m) * DD + 2 * lh;   // A: lane holds K pair at k0+2*lh
  v8f acc = {};
#pragma unroll
  for (int it = 0; it < 64; ++it) {
    v2f a = *(const v2f*)(arow + it * 4);
    v2f b = sB[it * 32 + lane];
    acc = __builtin_amdgcn_wmma_f32_16x16x4_f32(false, a, false, b, (short)0, acc, false, false);
  }
  if (m < 8) {                                    // cols 0..7 = heads, 8..15 = padding
    float bkv = s_bks[m];
    float* sc = ws + OFF_SCORES;
    int nb = tile * 16 + 8 * lh;
#pragma unroll
    for (int r = 0; r < 8; ++r)
      sc[(nb + r) * 8 + m] = acc[r] + bkv;        // D row = r + 8*lh
  }
}

// ---------------- K2: per-graph segment max / sum-exp; zero pooledX -----------
__global__ void k2_segred(const int* __restrict__ batch, float* __restrict__ ws) {
  int b = blockIdx.x, t = threadIdx.x;
  int lo = 0, hi = NN;
  while (lo < hi) { int mid = (lo + hi) >> 1; if (batch[mid] < b) lo = mid + 1; else hi = mid; }
  int start = lo;
  hi = NN;
  while (lo < hi) { int mid = (lo + hi) >> 1; if (batch[mid] < b + 1) lo = mid + 1; else hi = mid; }
  int end = lo;
  int h = t & 7, g = t >> 3;
  const float* sc = ws + OFF_SCORES;
  __shared__ float red[256];
  __shared__ float smax[8];
  float m = -__builtin_inff();
  for (int n = start + g; n < end; n += 32) m = fmaxf(m, sc[n * 8 + h]);
  red[t] = m; __syncthreads();
  for (int s = 128; s >= 8; s >>= 1) { if (t < s) red[t] = fmaxf(red[t], red[t + s]); __syncthreads(); }
  if (t < 8) smax[t] = red[t];
  __syncthreads();
  float segm = smax[h];
  float ssum = 0.f;
  for (int n = start + g; n < end; n += 32) ssum += __expf(sc[n * 8 + h] - segm);
  red[t] = ssum; __syncthreads();
  for (int s = 128; s >= 8; s >>= 1) { if (t < s) red[t] += red[t + s]; __syncthreads(); }
  if (t < 8) {
    ws[OFF_SEGMAX + b * 8 + t] = smax[t];
    float dn = red[t];
    ws[OFF_INV + b * 8 + t] = dn > 0.f ? 1.f / dn : 1.f;
  }
  if (t == 0) ws[OFF_FLAG + b] = (red[0] > 0.f) ? 1.f : 0.f;
  for (int i = t; i < HH * DD; i += 256) ws[OFF_POOLEDX + b * (HH * DD) + i] = 0.f;
}

// ---------------- K3: pooledX[b][h][d] += exp(score - max) * x[n][d] ----------
__global__ void k3_poolx(const float* __restrict__ x, const int* __restrict__ batch,
                         float* __restrict__ ws) {
  __shared__ __align__(16) float s_w[32 * 8];
  __shared__ int s_b[32];
  int t = threadIdx.x;                            // thread owns column d = t
  int w0 = blockIdx.x * WIN;
  int wEnd = min(w0 + WIN, NN);
  const float* sc = ws + OFF_SCORES;
  float* px = ws + OFF_POOLEDX;
  float acc[8] = {0.f, 0.f, 0.f, 0.f, 0.f, 0.f, 0.f, 0.f};
  int cur = -1;
  int ni = t >> 3, hh = t & 7;
  for (int ns = w0; ns < wEnd; ns += 32) {
    int cnt = min(32, wEnd - ns);
    if (ni < cnt) {                               // stage exp-weights for 32 nodes
      int n = ns + ni;
      int bb = batch[n];
      if (hh == 0) s_b[ni] = bb;
      s_w[t] = __expf(sc[n * 8 + hh] - ws[OFF_SEGMAX + bb * 8 + hh]);
    }
    __syncthreads();
    for (int i = 0; i < cnt; ++i) {
      int bb = s_b[i];
      if (bb != cur) {                            // block-uniform: flush on segment change
        if (cur >= 0) {
#pragma unroll
          for (int h2 = 0; h2 < 8; ++h2) {
            atomicAdd(&px[(cur * 8 + h2) * DD + t], acc[h2]);
            acc[h2] = 0.f;
          }
        }
        cur = bb;
      }
      float xv = x[(ns + i) * DD + t];
      const float4* wp = (const float4*)&s_w[i * 8];
      float4 wa = wp[0], wb = wp[1];
      acc[0] += wa.x * xv; acc[1] += wa.y * xv; acc[2] += wa.z * xv; acc[3] += wa.w * xv;
      acc[4] += wb.x * xv; acc[5] += wb.y * xv; acc[6] += wb.z * xv; acc[7] += wb.w * xv;
    }
    __syncthreads();
  }
  if (cur >= 0) {
#pragma unroll
    for (int h2 = 0; h2 < 8; ++h2)
      atomicAdd(&px[(cur * 8 + h2) * DD + t], acc[h2]);
  }
}

// ---------------- K4a: pooled = (pooledX @ Wv_head) * inv + bv  (WMMA) --------
__global__ void k4a_pool(const float* __restrict__ Wv, const float* __restrict__ bv,
                         float* __restrict__ ws) {
  int t = threadIdx.x;
  int h = blockIdx.x;                             // head
  int mt = t >> 5;                                // wave -> 16-row tile of B(=128)
  int lane = t & 31, m = lane & 15, lh = lane >> 4;
  const float* px = ws + OFF_POOLEDX;
  const float* inv = ws + OFF_INV;
  float* pooled = ws + OFF_POOLED;
  const float* aptr = px + ((mt * 16 + m) * 8 + h) * DD + 2 * lh;
  for (int nt = 0; nt < 2; ++nt) {
    v8f acc = {};
    int c = h * 32 + nt * 16 + m;
#pragma unroll
    for (int it = 0; it < 64; ++it) {
      int k0 = it * 4;
      v2f a = *(const v2f*)(aptr + k0);
      v2f bvec;
      const float* bp = Wv + (k0 + 2 * lh) * DD + c;
      bvec.x = bp[0];
      bvec.y = bp[DD];
      acc = __builtin_amdgcn_wmma_f32_16x16x4_f32(false, a, false, bvec, (short)0, acc, false, false);
    }
    float bvc = bv[c];
#pragma unroll
    for (int r = 0; r < 8; ++r) {
      int row = mt * 16 + r + 8 * lh;
      pooled[row * DD + c] = acc[r] * inv[row * 8 + h] + bvc;
    }
  }
}

// ---------------- K4b: out = mask * (pooled @ Wo + bo)  (WMMA) ----------------
__global__ void k4b_out(const float* __restrict__ Wo, const float* __restrict__ bo,
                        const float* __restrict__ ws, float* __restrict__ out) {
  int t = threadIdx.x;
  int id = blockIdx.x * 8 + (t >> 5);             // 128 (mt,nt) tiles
  int mt = id >> 4, nt = id & 15;
  int lane = t & 31, m = lane & 15, lh = lane >> 4;
  const float* pooled = ws + OFF_POOLED;
  const float* flag = ws + OFF_FLAG;
  const float* aptr = pooled + (mt * 16 + m) * DD + 2 * lh;
  v8f acc = {};
  int c = nt * 16 + m;
#pragma unroll
  for (int it = 0; it < 64; ++it) {
    int k0 = it * 4;
    v2f a = *(const v2f*)(aptr + k0);
    v2f bvec;
    const float* bp = Wo + (k0 + 2 * lh) * DD + c;
    bvec.x = bp[0];
    bvec.y = bp[DD];
    acc = __builtin_amdgcn_wmma_f32_16x16x4_f32(false, a, false, bvec, (short)0, acc, false, false);
  }
  float boc = bo[c];
#pragma unroll
  for (int r = 0; r < 8; ++r) {
    int row = mt * 16 + r + 8 * lh;
    out[row * DD + c] = flag[row] * (acc[r] + boc);
  }
}

extern "C" void kernel_launch(void* const* d_in, const int* in_sizes, int n_in,
                              void* d_out, int out_size, void* d_ws, size_t ws_size,
                              hipStream_t stream) {
  (void)in_sizes; (void)n_in; (void)out_size; (void)ws_size;
  const float* x   = (const float*)d_in[0];
  const int*   bat = (const int*)d_in[1];
  const float* gt  = (const float*)d_in[2];
  const float* Wq  = (const float*)d_in[3];
  const float* bq  = (const float*)d_in[4];
  const float* Wk  = (const float*)d_in[5];
  const float* bk  = (const float*)d_in[6];
  const float* Wv  = (const float*)d_in[7];
  const float* bv  = (const float*)d_in[8];
  const float* Wo  = (const float*)d_in[9];
  const float* bo  = (const float*)d_in[10];
  float* out = (float*)d_out;
  float* ws  = (float*)d_ws;

  k0_prep  <<<1, 256, 0, stream>>>(gt, Wq, bq, Wk, bk, ws);
  k1_scores<<<(NN / 16 + 7) / 8, 256, 0, stream>>>(x, ws);      // 31250 tiles / 8 waves
  k2_segred<<<BB, 256, 0, stream>>>(bat, ws);
  k3_poolx <<<(NN + WIN - 1) / WIN, 256, 0, stream>>>(x, bat, ws);
  k4a_pool <<<HH, 256, 0, stream>>>(Wv, bv, ws);
  k4b_out  <<<16, 256, 0, stream>>>(Wo, bo, ws, out);
}